// Seq2Seq_50611894616755
// MI455X (gfx1250) — compile-verified
//
#include <hip/hip_runtime.h>
#include <math.h>

// ---------------------------------------------------------------------------
// Seq2Seq transformer (enc-dec, post-LN) + KAN head, fp32, MI455X/gfx1250.
// All matmuls via V_WMMA_F32_16X16X4_F32 (fp32 in/out — matches reference
// numerics, no conversion traffic; workload is bandwidth/latency bound).
// Wave32 everywhere. Each wave computes a 16x64 output strip (4 accumulators)
// so the A fragment is amortized over 4 WMMAs: 5 b64 loads : 4 wmma.
// ---------------------------------------------------------------------------

typedef __attribute__((ext_vector_type(2))) float v2f;
typedef __attribute__((ext_vector_type(8))) float v8f;

constexpr int Bc  = 4;
constexpr int Sc  = 256;     // SRC_LEN == TRG_LEN
constexpr int Dc  = 512;
constexpr int HDc = 64;
constexpr int DFFc = 2048;
constexpr int Vocab = 32000;
constexpr int MR  = Bc * Sc; // 1024 rows per stream

static __device__ __forceinline__ float wave_sum32(float v) {
  #pragma unroll
  for (int o = 16; o; o >>= 1) v += __shfl_xor(v, o, 32);
  return v;
}
static __device__ __forceinline__ float wave_max32(float v) {
  #pragma unroll
  for (int o = 16; o; o >>= 1) v = fmaxf(v, __shfl_xor(v, o, 32));
  return v;
}

static __device__ __forceinline__ v8f wmma4(v2f a, v2f b, v8f c) {
  return __builtin_amdgcn_wmma_f32_16x16x4_f32(false, a, false, b,
                                               (short)0, c, false, false);
}

// ---------------------------------------------------------------------------
// fp32 WMMA GEMM: C[M,N] = A[M,K] @ B.  BT=0: B is [K,N] row-major.
// BT=1: B is [N,K] row-major (C = A @ B^T).  RELU applied on store.
// Block = 256 threads = 8 waves arranged 2(M) x 4(N); each wave owns a 16x64
// strip -> block tile 32x256.  Requires M%32==0, N%256==0 (true for all uses).
// A fragment (16x4 f32): lanes 0-15 hold K=k0,k0+1 ; lanes 16-31 K=k0+2,k0+3.
// B fragment (4x16 f32): lane = N column, same K half-split.
// C/D: VGPR i -> row m0 + i + 8*(lane>=16), col n0 + (lane&15).
// ---------------------------------------------------------------------------
template <int BT, int RELU>
__global__ void gemm_wmma(const float* __restrict__ A, const float* __restrict__ Bm,
                          float* __restrict__ C, int M, int N, int K,
                          int lda, int ldb, int ldc) {
  const int lane = threadIdx.x & 31;
  const int wave = threadIdx.x >> 5;
  const int m0 = blockIdx.y * 32 + (wave >> 2) * 16;
  const int n0 = blockIdx.x * 256 + (wave & 3) * 64;
  if (m0 >= M || n0 >= N) return;
  const int half = lane >> 4;   // 0: K=k0,k0+1   1: K=k0+2,k0+3
  const int l16  = lane & 15;
  const size_t aRow = (size_t)(m0 + l16) * lda;
  v8f ac0 = {}, ac1 = {}, ac2 = {}, ac3 = {};
  #pragma unroll 4
  for (int k0 = 0; k0 < K; k0 += 4) {
    const int ka = k0 + half * 2;
    v2f a;
    a.x = A[aRow + ka];
    a.y = A[aRow + ka + 1];
    v2f b0, b1, b2, b3;
    if (BT) {
      const float* Bp = Bm + (size_t)(n0 + l16) * ldb + ka;
      b0.x = Bp[0];              b0.y = Bp[1];
      b1.x = Bp[(size_t)16*ldb]; b1.y = Bp[(size_t)16*ldb + 1];
      b2.x = Bp[(size_t)32*ldb]; b2.y = Bp[(size_t)32*ldb + 1];
      b3.x = Bp[(size_t)48*ldb]; b3.y = Bp[(size_t)48*ldb + 1];
    } else {
      const float* Bp = Bm + (size_t)ka * ldb + n0 + l16;
      b0.x = Bp[0];   b0.y = Bp[ldb];
      b1.x = Bp[16];  b1.y = Bp[ldb + 16];
      b2.x = Bp[32];  b2.y = Bp[ldb + 32];
      b3.x = Bp[48];  b3.y = Bp[ldb + 48];
    }
    ac0 = wmma4(a, b0, ac0);
    ac1 = wmma4(a, b1, ac1);
    ac2 = wmma4(a, b2, ac2);
    ac3 = wmma4(a, b3, ac3);
  }
  const int rb = m0 + half * 8;
  #pragma unroll
  for (int i = 0; i < 8; ++i) {
    float v0 = ac0[i], v1 = ac1[i], v2 = ac2[i], v3 = ac3[i];
    if (RELU) {
      v0 = fmaxf(v0, 0.f); v1 = fmaxf(v1, 0.f);
      v2 = fmaxf(v2, 0.f); v3 = fmaxf(v3, 0.f);
    }
    float* Cp = C + (size_t)(rb + i) * ldc + n0 + l16;
    Cp[0] = v0; Cp[16] = v1; Cp[32] = v2; Cp[48] = v3;
  }
}

// ---------------------------------------------------------------------------
// Attention scores: S[bh,q,n] = (Q_bh[q,:] . K_bh[n,:]) / 8, masked -> -1e9.
// MODE 0: key pad mask (tok[b*S+n]!=0).  MODE 1: query pad mask & causal.
// grid (32 bh, 8), block 256; wave -> 16(q) x 64(n) strip; K-dim = 64.
// ---------------------------------------------------------------------------
template <int MODE>
__global__ void attn_scores(const float* __restrict__ Q, const float* __restrict__ Kmat,
                            float* __restrict__ Sc_out, const int* __restrict__ tok) {
  const int lane = threadIdx.x & 31;
  const int wave = threadIdx.x >> 5;
  const int bh = blockIdx.x;            // 0..31
  const int b  = bh >> 3;
  const int h  = bh & 7;
  const int tile = blockIdx.y * 8 + wave;   // 0..63
  const int q0 = (tile >> 2) * 16;
  const int n0 = (tile & 3) * 64;
  const int half = lane >> 4;
  const int l16  = lane & 15;
  const float* Qb = Q    + (size_t)b * Sc * Dc + h * HDc + (size_t)(q0 + l16) * Dc;
  const float* Kb = Kmat + (size_t)b * Sc * Dc + h * HDc + (size_t)(n0 + l16) * Dc;
  v8f ac0 = {}, ac1 = {}, ac2 = {}, ac3 = {};
  #pragma unroll
  for (int k0 = 0; k0 < HDc; k0 += 4) {
    const int ka = k0 + half * 2;
    v2f a;  a.x = Qb[ka];                    a.y = Qb[ka + 1];
    v2f b0; b0.x = Kb[ka];                   b0.y = Kb[ka + 1];
    v2f b1; b1.x = Kb[(size_t)16*Dc + ka];   b1.y = Kb[(size_t)16*Dc + ka + 1];
    v2f b2; b2.x = Kb[(size_t)32*Dc + ka];   b2.y = Kb[(size_t)32*Dc + ka + 1];
    v2f b3; b3.x = Kb[(size_t)48*Dc + ka];   b3.y = Kb[(size_t)48*Dc + ka + 1];
    ac0 = wmma4(a, b0, ac0);
    ac1 = wmma4(a, b1, ac1);
    ac2 = wmma4(a, b2, ac2);
    ac3 = wmma4(a, b3, ac3);
  }
  float* Sb = Sc_out + (size_t)bh * Sc * Sc;
  const float scale = 0.125f;            // 1/sqrt(64)
  #pragma unroll
  for (int i = 0; i < 8; ++i) {
    const int qq = q0 + half * 8 + i;
    float* Sp = Sb + (size_t)qq * Sc + n0 + l16;
    const float vv[4] = { ac0[i], ac1[i], ac2[i], ac3[i] };
    #pragma unroll
    for (int t = 0; t < 4; ++t) {
      const int nn = n0 + t * 16 + l16;
      bool ok;
      if (MODE == 0) ok = (tok[b * Sc + nn] != 0);
      else           ok = (tok[b * Sc + qq] != 0) && (nn <= qq);
      Sp[t * 16] = ok ? vv[t] * scale : -1e9f;
    }
  }
}

// Row softmax over 256 cols. grid 32 (bh), block 256 (8 waves, wave per row).
__global__ void attn_softmax(float* __restrict__ Sc_buf) {
  const int lane = threadIdx.x & 31;
  const int wave = threadIdx.x >> 5;
  float* Sb = Sc_buf + (size_t)blockIdx.x * Sc * Sc;
  for (int r = wave; r < Sc; r += 8) {
    float* row = Sb + (size_t)r * Sc;
    float v[8];
    float mx = -3.0e38f;
    #pragma unroll
    for (int j = 0; j < 8; ++j) { v[j] = row[lane + j * 32]; mx = fmaxf(mx, v[j]); }
    mx = wave_max32(mx);
    float s = 0.0f;
    #pragma unroll
    for (int j = 0; j < 8; ++j) { v[j] = __expf(v[j] - mx); s += v[j]; }
    s = wave_sum32(s);
    const float inv = 1.0f / s;
    #pragma unroll
    for (int j = 0; j < 8; ++j) row[lane + j * 32] = v[j] * inv;
  }
}

// O[b,q,h*64+n] = sum_k P[bh,q,k] * V[b,k,h*64+n].
// grid (32 bh, 2), block 256; wave -> 16(q) x 64(full head) strip; K = 256.
__global__ void attn_pv(const float* __restrict__ P, const float* __restrict__ Vmat,
                        float* __restrict__ O) {
  const int lane = threadIdx.x & 31;
  const int wave = threadIdx.x >> 5;
  const int bh = blockIdx.x;
  const int b  = bh >> 3;
  const int h  = bh & 7;
  const int q0 = (blockIdx.y * 8 + wave) * 16;   // 0..240
  const int half = lane >> 4;
  const int l16  = lane & 15;
  const float* Pb = P    + (size_t)bh * Sc * Sc + (size_t)(q0 + l16) * Sc;
  const float* Vb = Vmat + (size_t)b * Sc * Dc + h * HDc;
  v8f ac0 = {}, ac1 = {}, ac2 = {}, ac3 = {};
  #pragma unroll 4
  for (int k0 = 0; k0 < Sc; k0 += 4) {
    const int ka = k0 + half * 2;
    v2f a;  a.x = Pb[ka]; a.y = Pb[ka + 1];
    const float* Vp = Vb + (size_t)ka * Dc + l16;
    v2f b0; b0.x = Vp[0];   b0.y = Vp[Dc];
    v2f b1; b1.x = Vp[16];  b1.y = Vp[Dc + 16];
    v2f b2; b2.x = Vp[32];  b2.y = Vp[Dc + 32];
    v2f b3; b3.x = Vp[48];  b3.y = Vp[Dc + 48];
    ac0 = wmma4(a, b0, ac0);
    ac1 = wmma4(a, b1, ac1);
    ac2 = wmma4(a, b2, ac2);
    ac3 = wmma4(a, b3, ac3);
  }
  #pragma unroll
  for (int i = 0; i < 8; ++i) {
    const int qq = q0 + half * 8 + i;
    float* Op = O + (size_t)(b * Sc + qq) * Dc + h * HDc + l16;
    Op[0] = ac0[i]; Op[16] = ac1[i]; Op[32] = ac2[i]; Op[48] = ac3[i];
  }
}

// out = LayerNorm(x + d), row width 512. One wave per row; 8 rows per block.
__global__ void add_ln(const float* __restrict__ X, const float* __restrict__ Dl,
                       float* __restrict__ Out) {
  const int lane = threadIdx.x & 31;
  const int wave = threadIdx.x >> 5;
  const int row = blockIdx.x * 8 + wave;
  const size_t base = (size_t)row * Dc;
  float v[16];
  float s = 0.0f;
  #pragma unroll
  for (int j = 0; j < 16; ++j) {
    v[j] = X[base + lane + j * 32] + Dl[base + lane + j * 32];
    s += v[j];
  }
  const float mean = wave_sum32(s) * (1.0f / Dc);
  float var = 0.0f;
  #pragma unroll
  for (int j = 0; j < 16; ++j) { const float d = v[j] - mean; var += d * d; }
  var = wave_sum32(var) * (1.0f / Dc);
  const float inv = rsqrtf(var + 1e-5f);
  #pragma unroll
  for (int j = 0; j < 16; ++j) Out[base + lane + j * 32] = (v[j] - mean) * inv;
}

// x[row,:] = embed[tok[row],:] * sqrt(512)
__global__ void embed_scale(const int* __restrict__ tok, const float* __restrict__ E,
                            float* __restrict__ X) {
  const int row = blockIdx.x;
  const float* e = E + (size_t)tok[row] * Dc;
  const float sc = sqrtf((float)Dc);
  for (int j = threadIdx.x; j < Dc; j += blockDim.x)
    X[(size_t)row * Dc + j] = e[j] * sc;
}

// KAN inner: cubic B-spline basis (8 uniform knots on [-1,1]) of tanh(y),
// first 2 basis functions dotted with inner_coeffs[d, 0:2].
__global__ void kan_inner(const float* __restrict__ Y, const float* __restrict__ Cf,
                          float* __restrict__ Inner) {
  const int idx = blockIdx.x * blockDim.x + threadIdx.x;
  if (idx >= MR * Dc) return;
  const int d = idx & (Dc - 1);
  const float x = tanhf(Y[idx]);
  float t[8];
  #pragma unroll
  for (int i = 0; i < 8; ++i) t[i] = -1.0f + (2.0f / 7.0f) * (float)i;
  float b0[7];
  #pragma unroll
  for (int j = 0; j < 7; ++j) b0[j] = (t[j] <= x && x < t[j + 1]) ? 1.0f : 0.0f;
  float b1[6];
  #pragma unroll
  for (int j = 0; j < 6; ++j)
    b1[j] = (x - t[j]) / (t[j + 1] - t[j]) * b0[j] +
            (t[j + 2] - x) / (t[j + 2] - t[j + 1]) * b0[j + 1];
  float b2[5];
  #pragma unroll
  for (int j = 0; j < 5; ++j)
    b2[j] = (x - t[j]) / (t[j + 2] - t[j]) * b1[j] +
            (t[j + 3] - x) / (t[j + 3] - t[j + 1]) * b1[j + 1];
  float b3[4];
  #pragma unroll
  for (int j = 0; j < 4; ++j)
    b3[j] = (x - t[j]) / (t[j + 3] - t[j]) * b2[j] +
            (t[j + 4] - x) / (t[j + 4] - t[j + 1]) * b2[j + 1];
  Inner[idx] = b3[0] * Cf[d * 5 + 0] + b3[1] * Cf[d * 5 + 1];
}

// ---------------------------------------------------------------------------
extern "C" void kernel_launch(void* const* d_in, const int* in_sizes, int n_in,
                              void* d_out, int out_size, void* d_ws, size_t ws_size,
                              hipStream_t stream) {
  (void)in_sizes; (void)n_in; (void)out_size; (void)ws_size;
  const int*   src   = (const int*)d_in[0];
  const int*   trg   = (const int*)d_in[1];
  const float* embed = (const float*)d_in[2];
  const float* eWq = (const float*)d_in[3];
  const float* eWk = (const float*)d_in[4];
  const float* eWv = (const float*)d_in[5];
  const float* eWo = (const float*)d_in[6];
  const float* eW1 = (const float*)d_in[7];
  const float* eW2 = (const float*)d_in[8];
  const float* dWq  = (const float*)d_in[9];
  const float* dWk  = (const float*)d_in[10];
  const float* dWv  = (const float*)d_in[11];
  const float* dWo  = (const float*)d_in[12];
  const float* dWq2 = (const float*)d_in[13];
  const float* dWk2 = (const float*)d_in[14];
  const float* dWv2 = (const float*)d_in[15];
  const float* dWo2 = (const float*)d_in[16];
  const float* dW1  = (const float*)d_in[17];
  const float* dW2  = (const float*)d_in[18];
  const float* innerC = (const float*)d_in[19];
  const float* outerC = (const float*)d_in[20];
  float* out = (float*)d_out;

  // Workspace layout (fp32): 7 x [1024,512] + [1024,2048] + scores[32,256,256]
  float* ws = (float*)d_ws;
  const size_t SZ = (size_t)MR * Dc;            // 524288
  float* x  = ws + 0 * SZ;                      // encoder stream / enc memory
  float* y  = ws + 1 * SZ;                      // decoder stream
  float* q  = ws + 2 * SZ;
  float* k  = ws + 3 * SZ;
  float* v  = ws + 4 * SZ;
  float* a  = ws + 5 * SZ;                      // attn out / kan inner
  float* p  = ws + 6 * SZ;                      // projection temp
  float* hb = ws + 7 * SZ;                      // [1024,2048] ffn hidden
  float* sc = hb + (size_t)MR * DFFc;           // [32,256,256] scores

  const size_t DD = (size_t)Dc * Dc;
  const size_t DF = (size_t)Dc * DFFc;

  auto gemm = [&](const float* A, const float* Bm, float* C, int M, int N, int K,
                  int lda, int ldb, int ldc, bool bt, bool relu) {
    dim3 g(N / 256, M / 32), blk(256);
    if (bt)        gemm_wmma<1, 0><<<g, blk, 0, stream>>>(A, Bm, C, M, N, K, lda, ldb, ldc);
    else if (relu) gemm_wmma<0, 1><<<g, blk, 0, stream>>>(A, Bm, C, M, N, K, lda, ldb, ldc);
    else           gemm_wmma<0, 0><<<g, blk, 0, stream>>>(A, Bm, C, M, N, K, lda, ldb, ldc);
  };
  auto attention = [&](const float* Qm, const float* Km, const float* Vm, float* Om,
                       const int* tok, int mode) {
    if (mode == 0) attn_scores<0><<<dim3(32, 8), 256, 0, stream>>>(Qm, Km, sc, tok);
    else           attn_scores<1><<<dim3(32, 8), 256, 0, stream>>>(Qm, Km, sc, tok);
    attn_softmax<<<32, 256, 0, stream>>>(sc);
    attn_pv<<<dim3(32, 2), 256, 0, stream>>>(sc, Vm, Om);
  };

  // ------------------------------ encoder ---------------------------------
  embed_scale<<<MR, 128, 0, stream>>>(src, embed, x);
  for (int l = 0; l < 4; ++l) {
    gemm(x, eWq + l * DD, q, MR, Dc, Dc, Dc, Dc, Dc, false, false);
    gemm(x, eWk + l * DD, k, MR, Dc, Dc, Dc, Dc, Dc, false, false);
    gemm(x, eWv + l * DD, v, MR, Dc, Dc, Dc, Dc, Dc, false, false);
    attention(q, k, v, a, src, 0);
    gemm(a, eWo + l * DD, p, MR, Dc, Dc, Dc, Dc, Dc, false, false);
    add_ln<<<MR / 8, 256, 0, stream>>>(x, p, x);
    gemm(x, eW1 + l * DF, hb, MR, DFFc, Dc, Dc, DFFc, DFFc, false, true);
    gemm(hb, eW2 + l * DF, p, MR, Dc, DFFc, DFFc, Dc, Dc, false, false);
    add_ln<<<MR / 8, 256, 0, stream>>>(x, p, x);
  }

  // ------------------------------ decoder ---------------------------------
  embed_scale<<<MR, 128, 0, stream>>>(trg, embed, y);
  for (int l = 0; l < 4; ++l) {
    // masked self-attention
    gemm(y, dWq + l * DD, q, MR, Dc, Dc, Dc, Dc, Dc, false, false);
    gemm(y, dWk + l * DD, k, MR, Dc, Dc, Dc, Dc, Dc, false, false);
    gemm(y, dWv + l * DD, v, MR, Dc, Dc, Dc, Dc, Dc, false, false);
    attention(q, k, v, a, trg, 1);
    gemm(a, dWo + l * DD, p, MR, Dc, Dc, Dc, Dc, Dc, false, false);
    add_ln<<<MR / 8, 256, 0, stream>>>(y, p, y);
    // cross-attention (K/V from encoder memory x)
    gemm(y, dWq2 + l * DD, q, MR, Dc, Dc, Dc, Dc, Dc, false, false);
    gemm(x, dWk2 + l * DD, k, MR, Dc, Dc, Dc, Dc, Dc, false, false);
    gemm(x, dWv2 + l * DD, v, MR, Dc, Dc, Dc, Dc, Dc, false, false);
    attention(q, k, v, a, src, 0);
    gemm(a, dWo2 + l * DD, p, MR, Dc, Dc, Dc, Dc, Dc, false, false);
    add_ln<<<MR / 8, 256, 0, stream>>>(y, p, y);
    // FFN
    gemm(y, dW1 + l * DF, hb, MR, DFFc, Dc, Dc, DFFc, DFFc, false, true);
    gemm(hb, dW2 + l * DF, p, MR, Dc, DFFc, DFFc, Dc, Dc, false, false);
    add_ln<<<MR / 8, 256, 0, stream>>>(y, p, y);
  }

  // ------------------------------ KAN head --------------------------------
  kan_inner<<<(MR * Dc) / 256, 256, 0, stream>>>(y, innerC, a);
  // out[1024, 32000] = inner @ outer^T   (outer_coeffs is [32000, 512])
  gemm(a, outerC, out, MR, Vocab, Dc, Dc, Dc, Vocab, true, false);
}